// BahdanauAttention_10866267259508
// MI455X (gfx1250) — compile-verified
//
#include <hip/hip_runtime.h>
#include <hip/hip_bf16.h>

// Problem constants (from reference)
#define BB 4
#define TQ 256
#define TV 256
#define HH 512
#define DD 512
#define UU 256

typedef float v2f __attribute__((ext_vector_type(2)));
typedef float v8f __attribute__((ext_vector_type(8)));

// ---------------------------------------------------------------------------
// Fast tanh: gfx1250 has V_TANH_F32 in the transcendental pipe (co-executes
// with VALU). Prefer the dedicated builtin; fall back to libm lowering.
// ---------------------------------------------------------------------------
__device__ __forceinline__ float fast_tanh(float x) {
#if __has_builtin(__builtin_amdgcn_tanhf)
  return __builtin_amdgcn_tanhf(x);
#else
  return tanhf(x);
#endif
}

// ---------------------------------------------------------------------------
// FP32 WMMA GEMM: Out[M,N] = X[M,K] @ W[K,N] (+ bias), row-major everywhere.
// One wave (32 lanes) computes one 16x16 output tile via V_WMMA_F32_16X16X4_F32.
// gridDim.y = batch; per-batch strides in elements.
//
// VGPR layouts (CDNA5 ISA 7.12.2):
//  A (16x4, MxK): lane L -> M = L&15, Kbase = (L>>4)*2; a[0]=A[M][Kbase], a[1]=A[M][Kbase+1]
//  B (4x16, KxN): lane L -> N = L&15; b[0]=B[Kbase][N], b[1]=B[Kbase+1][N]
//  C/D (16x16):   lane L, vgpr i -> M = i + (L>>4)*8, N = L&15
// ---------------------------------------------------------------------------
__global__ __launch_bounds__(32)
void wmma_gemm_f32(const float* __restrict__ Xb, const float* __restrict__ Wb,
                   const float* __restrict__ bias, float* __restrict__ Ob,
                   int M, int N, int K,
                   long xStride, long wStride, long oStride)
{
  const float* X = Xb + (long)blockIdx.y * xStride;
  const float* W = Wb + (long)blockIdx.y * wStride;
  float*       O = Ob + (long)blockIdx.y * oStride;

  const int ntn = N >> 4;
  const int m0 = (blockIdx.x / ntn) << 4;
  const int n0 = (blockIdx.x % ntn) << 4;

  const int lane = threadIdx.x;
  const int half = lane >> 4;      // 0 or 1
  const int l15  = lane & 15;

  v8f acc = {};

  // A: two consecutive K-values per lane -> one 8-byte (b64) load per step
  const float* arow = X + (long)(m0 + l15) * K + (half << 1);
  // B: two rows (Kbase, Kbase+1) at column n0+l15
  const float* bcol = W + (long)(half << 1) * N + (n0 + l15);

  for (int k = 0; k < K; k += 4) {
    v2f a = *(const v2f*)arow;            // A[M][Kbase], A[M][Kbase+1]
    v2f b;
    b[0] = bcol[0];                       // B[Kbase][N]
    b[1] = bcol[N];                       // B[Kbase+1][N]
    acc = __builtin_amdgcn_wmma_f32_16x16x4_f32(
        /*neg_a=*/false, a, /*neg_b=*/false, b,
        /*c_mod=*/(short)0, acc, /*reuse_a=*/false, /*reuse_b=*/false);
    arow += 4;
    bcol += 4L * N;
  }

  const float bv = bias ? bias[n0 + l15] : 0.0f;
  const int mbase = m0 + (half << 3);
#pragma unroll
  for (int i = 0; i < 8; ++i)
    O[(long)(mbase + i) * N + n0 + l15] = acc[i] + bv;
}

// ---------------------------------------------------------------------------
// Fused score + softmax. One 256-thread block per (b, q) row; thread t owns
// score[b,q,t]:  acc = sum_u tanh(qp[u] + vp[b,t,u]) * V[u];  then a block
// softmax over the 256 scores. Writes attn[b,q,:] (the 2nd reference output).
// The 256-float q-row and V live in LDS; vp rows stream as float4 (L2-resident).
// ---------------------------------------------------------------------------
__global__ __launch_bounds__(256)
void score_softmax(const float* __restrict__ qp,   // [B*TQ, U]
                   const float* __restrict__ vp,   // [B*TV, U]
                   const float* __restrict__ Vvec, // [U]
                   const float* __restrict__ bV,   // [1]
                   float* __restrict__ attn)       // [B*TQ, TV]
{
  __shared__ float qrow[UU];
  __shared__ float vv[UU];
  __shared__ float red[TV];

  const int row = blockIdx.x;      // b*TQ + q
  const int b   = row >> 8;        // TQ == 256
  const int t   = threadIdx.x;     // tv index

  qrow[t] = qp[(long)row * UU + t];
  vv[t]   = Vvec[t];
  __syncthreads();

  const float4* vr = (const float4*)(vp + ((long)(b * TV + t)) * UU);
  float acc = 0.0f;
#pragma unroll 4
  for (int u4 = 0; u4 < UU / 4; ++u4) {
    float4 v = vr[u4];
    int u = u4 << 2;
    acc += fast_tanh(qrow[u + 0] + v.x) * vv[u + 0];
    acc += fast_tanh(qrow[u + 1] + v.y) * vv[u + 1];
    acc += fast_tanh(qrow[u + 2] + v.z) * vv[u + 2];
    acc += fast_tanh(qrow[u + 3] + v.w) * vv[u + 3];
  }
  float score = acc + bV[0];

  // block softmax over TV=256
  red[t] = score;
  __syncthreads();
  for (int s = 128; s > 0; s >>= 1) {
    if (t < s) red[t] = fmaxf(red[t], red[t + s]);
    __syncthreads();
  }
  const float mx = red[0];
  __syncthreads();
  const float e = __expf(score - mx);
  red[t] = e;
  __syncthreads();
  for (int s = 128; s > 0; s >>= 1) {
    if (t < s) red[t] += red[t + s];
    __syncthreads();
  }
  attn[(long)row * TV + t] = e * (1.0f / red[0]);
}

// ---------------------------------------------------------------------------
// kernel_launch
//   d_in: query, values, W1, b1, W2, b2, V, bV
//   d_out: context [B,TQ,D] (524288 f32) then attn [B,TQ,TV,1] (262144 f32)
//   d_ws : q_proj [B*TQ,U] (1MB) + v_proj [B*TV,U] (1MB)
// ---------------------------------------------------------------------------
extern "C" void kernel_launch(void* const* d_in, const int* in_sizes, int n_in,
                              void* d_out, int out_size, void* d_ws, size_t ws_size,
                              hipStream_t stream) {
  const float* query  = (const float*)d_in[0];  // [B,TQ,H]
  const float* values = (const float*)d_in[1];  // [B,TV,D]
  const float* W1     = (const float*)d_in[2];  // [H,U]
  const float* b1     = (const float*)d_in[3];  // [U]
  const float* W2     = (const float*)d_in[4];  // [D,U]
  const float* b2     = (const float*)d_in[5];  // [U]
  const float* Vv     = (const float*)d_in[6];  // [U,1]
  const float* bV     = (const float*)d_in[7];  // [1]

  float* context = (float*)d_out;                          // [B*TQ, D]
  float* attn    = (float*)d_out + (long)BB * TQ * DD;     // [B*TQ, TV]

  float* q_proj = (float*)d_ws;                            // [B*TQ, U]
  float* v_proj = q_proj + (long)BB * TQ * UU;             // [B*TV, U]

  // 1) q_proj = query @ W1 + b1   (M=1024, N=256, K=512)
  {
    dim3 grid((BB * TQ / 16) * (UU / 16), 1);
    wmma_gemm_f32<<<grid, 32, 0, stream>>>(query, W1, b1, q_proj,
                                           BB * TQ, UU, HH, 0, 0, 0);
  }
  // 2) v_proj = values @ W2 + b2  (M=1024, N=256, K=512)
  {
    dim3 grid((BB * TV / 16) * (UU / 16), 1);
    wmma_gemm_f32<<<grid, 32, 0, stream>>>(values, W2, b2, v_proj,
                                           BB * TV, UU, DD, 0, 0, 0);
  }
  // 3) fused score + softmax -> attn (in d_out)
  score_softmax<<<BB * TQ, 256, 0, stream>>>(q_proj, v_proj, Vv, bV, attn);

  // 4) context[b] = attn[b] @ values[b]   (batched: M=256, N=512, K=256)
  {
    dim3 grid((TQ / 16) * (DD / 16), BB);
    wmma_gemm_f32<<<grid, 32, 0, stream>>>(attn, values, nullptr, context,
                                           TQ, DD, TV,
                                           (long)TQ * TV, (long)TV * DD, (long)TQ * DD);
  }
}